// WbwAttentionLayer_6339371729295
// MI455X (gfx1250) — compile-verified
//
#include <hip/hip_runtime.h>
#include <hip/hip_bf16.h>
#include <math.h>

typedef __attribute__((ext_vector_type(2))) float v2f;
typedef __attribute__((ext_vector_type(8))) float v8f;

#define B_    32
#define LPREM 256
#define THYP  128
#define DDIM  256
#define SEQ   (LPREM + THYP)   // 384

#if __has_builtin(__builtin_amdgcn_tanhf)
#define FAST_TANH(x) __builtin_amdgcn_tanhf(x)
#else
#define FAST_TANH(x) tanhf(x)
#endif

// One workgroup per batch element. 256 threads = 8 waves (wave32).
// Prologue (all WMMA, V_WMMA_F32_16X16X4_F32):
//   WY[b] = premise[b] @ w_y  -> LDS (256 KB, fits the 320 KB WGP LDS)
//   SH[b] = hyp[b]     @ w_h  -> d_ws (removes h@w_h from the recurrence)
// Then 128 sequential recurrence steps entirely within the workgroup.
__launch_bounds__(256, 1)
__global__ void wbw_attention_kernel(const float* __restrict__ x,
                                     const float* __restrict__ w_y,
                                     const float* __restrict__ w_h,
                                     const float* __restrict__ w_r,
                                     const float* __restrict__ w_alpha,
                                     const float* __restrict__ w_t,
                                     float* __restrict__ sh_ws,
                                     float* __restrict__ out)
{
    extern __shared__ float smem[];
    float* WY    = smem;                  // LPREM*DDIM = 65536 floats (256 KB)
    float* s_sh  = WY + LPREM * DDIM;     // DDIM
    float* r_sh  = s_sh + DDIM;           // DDIM
    float* alpha = r_sh + DDIM;           // LPREM (scores, then softmax weights)
    float* red   = alpha + LPREM;         // 16 (block reductions)

    const int b    = blockIdx.x;
    const int tid  = threadIdx.x;
    const int lane = tid & 31;
    const int wave = tid >> 5;

    const float* xb  = x + (size_t)b * SEQ * DDIM; // premise row l: xb[l*D+d]; hyp row t: xb[(L+t)*D+d]
    float*       SHb = sh_ws + (size_t)b * THYP * DDIM;

    // ---------------- Prologue: two GEMMs via fp32 WMMA 16x16x4 ----------------
    // Lane mapping per CDNA5 ISA 7.12.2 (32-bit A 16x4 / B 4x16 / C 16x16 layouts).
    const int kbase = (lane >> 4) << 1;   // 0 or 2
    const int mrow  = lane & 15;          // A row within tile
    const int ncol  = lane & 15;          // B/C column within tile
    const int rsub  = (lane >> 4) << 3;   // C row offset: 0 or 8

    const int wyTiles = (LPREM / 16) * (DDIM / 16);  // 256
    const int shTiles = (THYP  / 16) * (DDIM / 16);  // 128
    for (int tIdx = wave; tIdx < wyTiles + shTiles; tIdx += 8) {
        const bool isWY = tIdx < wyTiles;
        const int  tt   = isWY ? tIdx : (tIdx - wyTiles);
        const int  tm   = tt >> 4;        // tile row
        const int  tn   = tt & 15;        // tile column
        const float* arow = isWY ? (xb + (tm * 16 + mrow) * DDIM)
                                 : (xb + (LPREM + tm * 16 + mrow) * DDIM);
        const float* W    = isWY ? w_y : w_h;
        v8f acc = {};
        #pragma unroll 4
        for (int k0 = 0; k0 < DDIM; k0 += 4) {
            v2f afrag, bfrag;
            afrag.x = arow[k0 + kbase];
            afrag.y = arow[k0 + kbase + 1];
            bfrag.x = W[(k0 + kbase)     * DDIM + tn * 16 + ncol];
            bfrag.y = W[(k0 + kbase + 1) * DDIM + tn * 16 + ncol];
            // 8 args: (neg_a, A, neg_b, B, c_mod, C, reuse_a, reuse_b)
            acc = __builtin_amdgcn_wmma_f32_16x16x4_f32(false, afrag, false, bfrag,
                                                        (short)0, acc, false, false);
        }
        const int rbase = tm * 16 + rsub; // C layout: VGPR v -> row v (lanes 0-15), v+8 (lanes 16-31)
        if (isWY) {
            #pragma unroll
            for (int v = 0; v < 8; ++v)
                WY[(rbase + v) * DDIM + tn * 16 + ncol] = acc[v];
        } else {
            #pragma unroll
            for (int v = 0; v < 8; ++v)
                SHb[(size_t)(rbase + v) * DDIM + tn * 16 + ncol] = acc[v];
        }
    }

    // init recurrent state; make SH stores visible to all waves in the workgroup
    r_sh[tid] = 0.0f;
    __threadfence();
    __syncthreads();

    // w_alpha slices are invariant across steps and rows: keep in registers
    float wa[DDIM / 32];
    #pragma unroll
    for (int m = 0; m < DDIM / 32; ++m) wa[m] = w_alpha[lane + 32 * m];

    // ---------------- Recurrence: 128 sequential steps ----------------
    for (int t = 0; t < THYP; ++t) {
        // s[d] = SH[t][d] + sum_k r[k]*w_r[k][d]   (thread d, coalesced weight reads)
        float sacc = SHb[(size_t)t * DDIM + tid];
        for (int k = 0; k < DDIM; ++k)
            sacc = fmaf(r_sh[k], w_r[k * DDIM + tid], sacc);
        s_sh[tid] = sacc;
        __syncthreads();

        // scores[l] = sum_d tanh(WY[l][d] + s[d]) * w_alpha[d]
        // wave-cooperative: lane = d-slice (bank-conflict-free LDS); hoist s into regs
        float sv[DDIM / 32];
        #pragma unroll
        for (int m = 0; m < DDIM / 32; ++m) sv[m] = s_sh[lane + 32 * m];

        for (int l = wave; l < LPREM; l += 8) {
            float p = 0.0f;
            #pragma unroll
            for (int m = 0; m < DDIM / 32; ++m) {
                const int d = lane + 32 * m;
                p = fmaf(FAST_TANH(WY[l * DDIM + d] + sv[m]), wa[m], p);
            }
            #pragma unroll
            for (int off = 16; off; off >>= 1) p += __shfl_xor(p, off, 32);
            if (lane == 0) alpha[l] = p;
        }
        __syncthreads();

        // softmax over L=256 scores
        float sc = alpha[tid];
        float mx = sc;
        #pragma unroll
        for (int off = 16; off; off >>= 1) mx = fmaxf(mx, __shfl_xor(mx, off, 32));
        if (lane == 0) red[wave] = mx;
        __syncthreads();
        if (tid == 0) {
            float mm = red[0];
            #pragma unroll
            for (int i = 1; i < 8; ++i) mm = fmaxf(mm, red[i]);
            red[8] = mm;
        }
        __syncthreads();
        const float e = __expf(sc - red[8]);
        float su = e;
        #pragma unroll
        for (int off = 16; off; off >>= 1) su += __shfl_xor(su, off, 32);
        if (lane == 0) red[wave] = su;
        __syncthreads();
        if (tid == 0) {
            float ss = 0.0f;
            #pragma unroll
            for (int i = 0; i < 8; ++i) ss += red[i];
            red[9] = ss;
        }
        __syncthreads();
        alpha[tid] = e / red[9];
        __syncthreads();

        // r_t[d] = sum_l alpha[l]*premise[l][d] + sum_k r[k]*w_t[k][d]
        float racc = 0.0f;
        for (int l = 0; l < LPREM; ++l)
            racc = fmaf(alpha[l], xb[l * DDIM + tid], racc);
        for (int k = 0; k < DDIM; ++k)
            racc = fmaf(r_sh[k], w_t[k * DDIM + tid], racc);
        __syncthreads();              // all reads of old r done
        r_sh[tid] = racc;
        out[((size_t)b * THYP + t) * DDIM + tid] = racc;
        __syncthreads();
    }
}

extern "C" void kernel_launch(void* const* d_in, const int* in_sizes, int n_in,
                              void* d_out, int out_size, void* d_ws, size_t ws_size,
                              hipStream_t stream) {
    (void)in_sizes; (void)n_in; (void)ws_size; (void)out_size;
    const float* x       = (const float*)d_in[0];
    const float* w_y     = (const float*)d_in[1];
    const float* w_h     = (const float*)d_in[2];
    const float* w_r     = (const float*)d_in[3];
    const float* w_alpha = (const float*)d_in[4];
    const float* w_t     = (const float*)d_in[5];
    // d_in[6] is L (==256), fixed at compile time
    float* out   = (float*)d_out;
    float* sh_ws = (float*)d_ws;   // B*T*D floats = 4 MB scratch for SH = hyp @ w_h

    const size_t shmem = (size_t)(LPREM * DDIM + 2 * DDIM + LPREM + 16) * sizeof(float);
    wbw_attention_kernel<<<B_, 256, shmem, stream>>>(x, w_y, w_h, w_r, w_alpha, w_t, sh_ws, out);
}